// GraphConvModulePureAutograd_86260123174005
// MI455X (gfx1250) — compile-verified
//
#include <hip/hip_runtime.h>
#include <hip/hip_bf16.h>

typedef __attribute__((ext_vector_type(2))) float v2f;
typedef __attribute__((ext_vector_type(8))) float v8f;

#define N_NODES   50000
#define N_FILTERS 4096
#define EDGE_FEAT 32
#define HIDDEN    128
#define IN_CH     16
#define OUT_CH    16

// ---------------------------------------------------------------- utilities
__global__ void zero_f32(float* __restrict__ p, int n) {
    int i = blockIdx.x * blockDim.x + threadIdx.x;
    if (i < n) p[i] = 0.0f;
}

// ------------------------------------------------------- 1) histogram of idxe
__global__ __launch_bounds__(256) void hist_kernel(const int* __restrict__ idxe,
                                                   int* __restrict__ counts, int n) {
    __shared__ int lc[N_FILTERS];
    for (int i = threadIdx.x; i < N_FILTERS; i += blockDim.x) lc[i] = 0;
    __syncthreads();
    int gid = blockIdx.x * blockDim.x + threadIdx.x;
    int stride = gridDim.x * blockDim.x;
    for (int e = gid; e < n; e += stride) atomicAdd(&lc[idxe[e]], 1);
    __syncthreads();
    for (int i = threadIdx.x; i < N_FILTERS; i += blockDim.x) {
        int v = lc[i];
        if (v) atomicAdd(&counts[i], v);
    }
}

// -------------------------------------------- 2) exclusive scan (single block)
__global__ __launch_bounds__(1024) void scan_kernel(const int* __restrict__ counts,
                                                    int* __restrict__ bin_start) {
    __shared__ int tsum[1024];
    int t = threadIdx.x;                 // 1024 threads x 4 bins = 4096
    int4 v = ((const int4*)counts)[t];
    int s = v.x + v.y + v.z + v.w;
    tsum[t] = s;
    __syncthreads();
    for (int off = 1; off < 1024; off <<= 1) {
        int add = (t >= off) ? tsum[t - off] : 0;
        __syncthreads();
        tsum[t] += add;
        __syncthreads();
    }
    int excl = tsum[t] - s;
    bin_start[4 * t + 0] = excl;
    bin_start[4 * t + 1] = excl + v.x;
    bin_start[4 * t + 2] = excl + v.x + v.y;
    bin_start[4 * t + 3] = excl + v.x + v.y + v.z;
    if (t == 1023) bin_start[N_FILTERS] = excl + s;
}

// ----------------------------------------------- 3) scatter edges into bins
__global__ __launch_bounds__(256) void scatter_kernel(const int* __restrict__ idxe,
                                                      const int* __restrict__ bin_start,
                                                      int* __restrict__ cursor,
                                                      int* __restrict__ binned, int n) {
    int e = blockIdx.x * blockDim.x + threadIdx.x;
    if (e < n) {
        int f = idxe[e];
        int p = atomicAdd(&cursor[f], 1);
        binned[bin_start[f] + p] = e;
    }
}

// ------------------- 4) filter network: wts = relu(ef@W1+b1)@W2+b2  (WMMA f32)
__global__ __launch_bounds__(128) void filternet_kernel(const float* __restrict__ ef,
                                                        const float* __restrict__ W1,
                                                        const float* __restrict__ b1,
                                                        const float* __restrict__ W2,
                                                        const float* __restrict__ b2,
                                                        float* __restrict__ wts) {
    __shared__ float hbuf[4][16 * HIDDEN];   // 4 waves * 8KB = 32KB
    int lane = threadIdx.x & 31;
    int wv = threadIdx.x >> 5;
    int tile = blockIdx.x * 4 + wv;          // 256 tiles of 16 filters
    int fb = tile * 16;
    int lo = lane & 15, hi = lane >> 4;
    float* h = hbuf[wv];

    // GEMM1: [16x32] @ [32x128] -> h [16x128], relu + bias
    for (int nt = 0; nt < HIDDEN / 16; ++nt) {
        int nb = nt * 16;
        v8f c = {};
        for (int step = 0; step < EDGE_FEAT / 4; ++step) {
            int k0 = step * 4;
            v2f a = *(const v2f*)(ef + (fb + lo) * EDGE_FEAT + k0 + 2 * hi);
            v2f b;
            b.x = W1[(k0 + 2 * hi + 0) * HIDDEN + nb + lo];
            b.y = W1[(k0 + 2 * hi + 1) * HIDDEN + nb + lo];
            c = __builtin_amdgcn_wmma_f32_16x16x4_f32(false, a, false, b,
                                                      (short)0, c, false, false);
        }
        float bias = b1[nb + lo];
#pragma unroll
        for (int r = 0; r < 8; ++r) {
            float v = c[r] + bias;
            h[(r + 8 * hi) * HIDDEN + nb + lo] = v > 0.0f ? v : 0.0f;
        }
    }
    __syncthreads();

    // GEMM2: h [16x128] @ [128x256] -> wts tile [16x256], + bias
    for (int ot = 0; ot < (IN_CH * OUT_CH) / 16; ++ot) {
        int nb = ot * 16;
        v8f c = {};
        for (int step = 0; step < HIDDEN / 4; ++step) {
            int k0 = step * 4;
            v2f a = *(const v2f*)(h + lo * HIDDEN + k0 + 2 * hi);
            v2f b;
            b.x = W2[(k0 + 2 * hi + 0) * (IN_CH * OUT_CH) + nb + lo];
            b.y = W2[(k0 + 2 * hi + 1) * (IN_CH * OUT_CH) + nb + lo];
            c = __builtin_amdgcn_wmma_f32_16x16x4_f32(false, a, false, b,
                                                      (short)0, c, false, false);
        }
        float bias = b2[nb + lo];
#pragma unroll
        for (int r = 0; r < 8; ++r)
            wts[(fb + r + 8 * hi) * (IN_CH * OUT_CH) + nb + lo] = c[r] + bias;
    }
}

// ---------------- 5a) per-bin GEMM (WMMA f32), store products at ORIGINAL
// edge index (coalesced 64B rows, L2-resident, no atomics -> deterministic)
__global__ __launch_bounds__(256) void edge_store_kernel(const float* __restrict__ input,
                                                         const int* __restrict__ idxn,
                                                         const float* __restrict__ wts,
                                                         const int* __restrict__ bin_start,
                                                         const int* __restrict__ binned,
                                                         float* __restrict__ products) {
    int f = blockIdx.x;                       // one block per filter bin
    int lane = threadIdx.x & 31;
    int wv = threadIdx.x >> 5;                // 8 waves
    int lo = lane & 15, hi = lane >> 4;
    int start = bin_start[f];
    int cnt = bin_start[f + 1] - start;
    if (cnt <= 0) return;

    // W_f in B-matrix layout: b[kk][r] = W_f[K = 4kk+2hi+r][N = lo]
    const float* Wf = wts + f * (IN_CH * OUT_CH);
    v2f b[4];
#pragma unroll
    for (int kk = 0; kk < 4; ++kk) {
        b[kk].x = Wf[(4 * kk + 2 * hi + 0) * OUT_CH + lo];
        b[kk].y = Wf[(4 * kk + 2 * hi + 1) * OUT_CH + lo];
    }

    int ntiles = (cnt + 15) >> 4;
    for (int t = wv; t < ntiles; t += 8) {
        int base = start + t * 16;
        int rem = cnt - t * 16; if (rem > 16) rem = 16;
        bool vm = lo < rem;
        int e   = vm ? binned[base + lo] : 0;
        int src = vm ? idxn[e] : 0;

        v8f c = {};
#pragma unroll
        for (int kk = 0; kk < 4; ++kk) {
            v2f a;
            if (vm) a = *(const v2f*)(input + src * IN_CH + 4 * kk + 2 * hi);
            else    { a.x = 0.0f; a.y = 0.0f; }
            c = __builtin_amdgcn_wmma_f32_16x16x4_f32(false, a, false, b[kk],
                                                      (short)0, c, false, false);
        }

        // D element in c[r] is products[row = r+8*hi][col = lo]
#pragma unroll
        for (int r = 0; r < 8; ++r) {
            int mrow = r + 8 * hi;
            int er = __shfl(e, mrow, 32);
            if (mrow < rem) products[er * OUT_CH + lo] = c[r];
        }
    }
}

// ---------------- 5b) fallback: fused atomic scatter variant (small ws)
__global__ __launch_bounds__(256) void edge_atomic_kernel(const float* __restrict__ input,
                                                          const int* __restrict__ idxn,
                                                          const int* __restrict__ seg,
                                                          const float* __restrict__ wts,
                                                          const int* __restrict__ bin_start,
                                                          const int* __restrict__ binned,
                                                          float* __restrict__ sums,
                                                          float* __restrict__ deg) {
    int f = blockIdx.x;
    int lane = threadIdx.x & 31;
    int wv = threadIdx.x >> 5;
    int lo = lane & 15, hi = lane >> 4;
    int start = bin_start[f];
    int cnt = bin_start[f + 1] - start;
    if (cnt <= 0) return;

    const float* Wf = wts + f * (IN_CH * OUT_CH);
    v2f b[4];
#pragma unroll
    for (int kk = 0; kk < 4; ++kk) {
        b[kk].x = Wf[(4 * kk + 2 * hi + 0) * OUT_CH + lo];
        b[kk].y = Wf[(4 * kk + 2 * hi + 1) * OUT_CH + lo];
    }

    int ntiles = (cnt + 15) >> 4;
    for (int t = wv; t < ntiles; t += 8) {
        int base = start + t * 16;
        int rem = cnt - t * 16; if (rem > 16) rem = 16;
        bool vm = lo < rem;
        int e   = vm ? binned[base + lo] : 0;
        int src = vm ? idxn[e] : 0;
        int nd  = vm ? seg[e] : 0;

        v8f c = {};
#pragma unroll
        for (int kk = 0; kk < 4; ++kk) {
            v2f a;
            if (vm) a = *(const v2f*)(input + src * IN_CH + 4 * kk + 2 * hi);
            else    { a.x = 0.0f; a.y = 0.0f; }
            c = __builtin_amdgcn_wmma_f32_16x16x4_f32(false, a, false, b[kk],
                                                      (short)0, c, false, false);
        }

        if (hi == 0 && vm) atomicAdd(deg + nd, 1.0f);
#pragma unroll
        for (int r = 0; r < 8; ++r) {
            int mrow = r + 8 * hi;
            int ndr = __shfl(nd, mrow, 32);
            if (mrow < rem) atomicAdd(sums + ndr * OUT_CH + lo, c[r]);
        }
    }
}

// ------------------- 6) per-node edge ranges via binary search (seg sorted)
__global__ void node_start_kernel(const int* __restrict__ seg,
                                  int* __restrict__ ns, int n_edges) {
    int n = blockIdx.x * blockDim.x + threadIdx.x;
    if (n <= N_NODES) {
        int lo = 0, hi = n_edges;          // first index with seg[idx] >= n
        while (lo < hi) {
            int mid = (lo + hi) >> 1;
            if (seg[mid] < n) lo = mid + 1; else hi = mid;
        }
        ns[n] = lo;
    }
}

// ------------------- 7) deterministic segment mean: one wave per node
__global__ __launch_bounds__(256) void segmean_kernel(const float* __restrict__ products,
                                                      const int* __restrict__ ns,
                                                      float* __restrict__ out) {
    int lane = threadIdx.x & 31;
    int wv = threadIdx.x >> 5;
    int n = blockIdx.x * 8 + wv;
    if (n >= N_NODES) return;
    int lo = lane & 15, hi = lane >> 4;
    int s = ns[n];
    int cnt = ns[n + 1] - s;
    float acc = 0.0f;
    for (int i = hi; i < cnt; i += 2)       // lanes cover 2 edges x 16 channels
        acc += products[(s + i) * OUT_CH + lo];
    acc += __shfl_xor(acc, 16, 32);         // combine the two edge slots
    if (hi == 0)
        out[n * OUT_CH + lo] = cnt > 0 ? acc / (float)cnt : 0.0f;
}

// --------------------------------------------- 8) fallback mean (atomic path)
__global__ void finalize_kernel(const float* __restrict__ sums,
                                const float* __restrict__ deg,
                                float* __restrict__ out, int n) {
    int i = blockIdx.x * blockDim.x + threadIdx.x;
    if (i < n) {
        float d = deg[i >> 4];
        out[i] = d > 0.0f ? sums[i] / d : 0.0f;
    }
}

// ---------------------------------------------------------------------------
extern "C" void kernel_launch(void* const* d_in, const int* in_sizes, int n_in,
                              void* d_out, int out_size, void* d_ws, size_t ws_size,
                              hipStream_t stream) {
    const float* input = (const float*)d_in[0];
    const int*   idxn  = (const int*)d_in[1];
    const int*   idxe  = (const int*)d_in[2];
    const int*   seg   = (const int*)d_in[3];
    const float* ef    = (const float*)d_in[4];
    const float* W1    = (const float*)d_in[5];
    const float* b1    = (const float*)d_in[6];
    const float* W2    = (const float*)d_in[7];
    const float* b2    = (const float*)d_in[8];
    float* out = (float*)d_out;

    const int n_edges = in_sizes[1];

    // workspace carve-out (256B aligned slabs)
    char* ws = (char*)d_ws;
    size_t off = 0;
    auto carve = [&](size_t bytes) -> void* {
        void* p = ws + off;
        off = (off + bytes + 255) & ~(size_t)255;
        return p;
    };
    int*   counts    = (int*)carve(N_FILTERS * sizeof(int));
    int*   cursor    = (int*)carve(N_FILTERS * sizeof(int));
    int*   bin_start = (int*)carve((N_FILTERS + 4) * sizeof(int));
    int*   binned    = (int*)carve((size_t)n_edges * sizeof(int));
    float* wts       = (float*)carve((size_t)N_FILTERS * IN_CH * OUT_CH * sizeof(float));
    size_t shared_off = off;

    // Path A (preferred): dense products buffer, L2-resident, no big atomics.
    size_t needA = shared_off + (((size_t)n_edges * OUT_CH * sizeof(float) + 255) & ~(size_t)255)
                              + (size_t)(N_NODES + 1) * sizeof(int) + 512;
    bool pathA = ws_size >= needA;

    // zero bin counters (fresh every call: deterministic)
    zero_f32<<<(N_FILTERS * 2 + 255) / 256, 256, 0, stream>>>((float*)counts, N_FILTERS * 2);

    // binning
    hist_kernel<<<512, 256, 0, stream>>>(idxe, counts, n_edges);
    scan_kernel<<<1, 1024, 0, stream>>>(counts, bin_start);
    scatter_kernel<<<(n_edges + 255) / 256, 256, 0, stream>>>(idxe, bin_start, cursor,
                                                              binned, n_edges);

    // dynamic filter weights via WMMA
    filternet_kernel<<<N_FILTERS / 64, 128, 0, stream>>>(ef, W1, b1, W2, b2, wts);

    if (pathA) {
        float* products = (float*)carve((size_t)n_edges * OUT_CH * sizeof(float));
        int*   ns       = (int*)carve((size_t)(N_NODES + 1) * sizeof(int));

        edge_store_kernel<<<N_FILTERS, 256, 0, stream>>>(input, idxn, wts, bin_start,
                                                         binned, products);
        node_start_kernel<<<(N_NODES + 1 + 255) / 256, 256, 0, stream>>>(seg, ns, n_edges);
        segmean_kernel<<<(N_NODES + 7) / 8, 256, 0, stream>>>(products, ns, out);
    } else {
        float* sums = (float*)carve((size_t)N_NODES * OUT_CH * sizeof(float));
        float* deg  = (float*)carve((size_t)N_NODES * sizeof(float));

        zero_f32<<<((N_NODES * OUT_CH) + 255) / 256, 256, 0, stream>>>(sums, N_NODES * OUT_CH);
        zero_f32<<<(N_NODES + 255) / 256, 256, 0, stream>>>(deg, N_NODES);
        edge_atomic_kernel<<<N_FILTERS, 256, 0, stream>>>(input, idxn, seg, wts, bin_start,
                                                          binned, sums, deg);
        finalize_kernel<<<((N_NODES * OUT_CH) + 255) / 256, 256, 0, stream>>>(
            sums, deg, out, N_NODES * OUT_CH);
    }
}